// CE_style_75342316306668
// MI455X (gfx1250) — compile-verified
//
#include <hip/hip_runtime.h>
#include <math.h>

// CDNA5 / gfx1250 SDDMM-style contrastive loss.
// Shapes: B=16, N=32, M=64, D=512 -> NM=2048, NUM_NEGS=32, T=1, smoothing=0.1.

typedef __attribute__((ext_vector_type(2))) float v2f;
typedef __attribute__((ext_vector_type(8))) float v8f;

#define B_   16
#define N_   32
#define M_   64
#define D_   512
#define NM_  2048
#define NEGS_ 32
#define SMOOTH_ 0.1f
#define QTILES_ (B_ * NM_ / 16)   // 2048 blocks

__global__ __launch_bounds__(256)
void ce_sddmm_wmma(const float* __restrict__ S,          // self_attd  (B,NM,D)
                   const float* __restrict__ C,          // cross_attd (B,NM,D)
                   const unsigned char* __restrict__ PAD, // padding_mask (B,N,M,M) bool
                   const unsigned char* __restrict__ VM,  // valid_negs_mask (B,NM,NM) bool
                   const int* __restrict__ NEG,           // sampled_neg_inds (B*NM*NEGS, 3) int32
                   float* __restrict__ parts)             // per-block partials [3]
{
    __shared__ float vals[16][34];   // [query-in-tile][j]; j=32 is the numerator
    __shared__ float red[3][16];

    const int blk = blockIdx.x;
    const int b   = blk >> 7;          // 128 q-tiles per batch
    const int q0  = (blk & 127) << 4;  // first query of this 16-row tile

    const int lane = threadIdx.x & 31;
    const int wave = threadIdx.x >> 5;
    const int r    = lane & 15;        // matrix row (A) / column (B) owned by this lane
    const int kh   = lane >> 4;        // K half: lanes 0-15 -> K{0,1}, lanes 16-31 -> K{2,3}
    const int q    = q0 + r;           // global query index for this lane's row

    const float* srow = S + ((size_t)b * NM_ + (size_t)q) * D_ + 2 * kh;

    // Each wave owns neg-slots j = wave, wave+8, ... ; wave 0 also does j=32 (numerator).
    for (int j = wave; j < 33; j += 8) {
        int col;
        if (j < 32) {
            // sampled_neg_inds row = b*NM*NEGS + q*NEGS + j, component 2 = column
            col = NEG[(((size_t)b * NM_ + (size_t)q) * NEGS_ + j) * 3 + 2];
        } else {
            col = q;   // numerator: diagonal dot s[q] . c[q]
        }
        const float* crow = C + ((size_t)b * NM_ + (size_t)col) * D_ + 2 * kh;

        // 128-step K chain: D=512 in K=4 slices. A = 16 queries, B cols = gathered c rows.
        v8f acc = {};
        #pragma unroll 8
        for (int t = 0; t < D_ / 4; ++t) {
            v2f av = *(const v2f*)(srow + 4 * t);   // A[row=r][K = 4t + 2*kh + {0,1}]
            v2f bv = *(const v2f*)(crow + 4 * t);   // B[K = 4t + 2*kh + {0,1}][col=r]
            acc = __builtin_amdgcn_wmma_f32_16x16x4_f32(
                false, av, false, bv, (short)0, acc, false, false);
        }

        // Diagonal extraction per C/D layout:
        // lanes 0-15: (M=v, N=lane); lanes 16-31: (M=8+v, N=lane-16).
        float a8[8];
        #pragma unroll
        for (int k = 0; k < 8; ++k) a8[k] = acc[k];

        int i = -1; float val = 0.f;
        if (lane < 8)        { i = lane;      val = a8[lane];      }   // M=N=lane
        else if (lane >= 24) { i = lane - 16; val = a8[lane - 24]; }   // M=N=lane-16

        if (i >= 0) {                 // note: i == r here, so q/col are consistent
            if (j < 32) {
                size_t midx = ((size_t)b * NM_ + (size_t)q) * (size_t)NM_ + (size_t)col;
                if (!VM[midx]) val = 0.f;
                vals[i][j] = val;
            } else {
                vals[i][32] = val;
            }
        }
    }

    __syncthreads();

    // Phase 2: per-query label-smoothed CE over [numerator, 32 negs] + sim loss.
    if (threadIdx.x < 16) {
        const int i = threadIdx.x;
        const float pos = vals[i][32];          // num / TEMPERATURE (T=1)
        float mx = pos;
        #pragma unroll
        for (int j = 0; j < 32; ++j) mx = fmaxf(mx, vals[i][j]);
        float se = expf(pos - mx);
        float ssum = 0.f;
        #pragma unroll
        for (int j = 0; j < 32; ++j) { se += expf(vals[i][j] - mx); ssum += vals[i][j]; }
        const float lse = logf(se) + mx;
        const float lp0 = pos - lse;
        const float sum_lp_negs = ssum - 32.f * lse;
        const float loss = -((1.f - SMOOTH_) * lp0 + (SMOOTH_ / 31.f) * sum_lp_negs);

        const int qg = q0 + i;
        const int n  = qg >> 6;        // M_=64
        const int mm = qg & 63;
        const size_t pidx = (((size_t)b * N_ + n) * M_ + mm) * (size_t)M_;  // [...,0]
        const float vf = PAD[pidx] ? 0.f : 1.f;
        const float sl = 1.f - fminf(fmaxf(pos, -1.f), 1.f);

        red[0][i] = loss * vf;
        red[1][i] = sl * vf;
        red[2][i] = vf;
    }
    __syncthreads();

    if (threadIdx.x == 0) {
        float ce = 0.f, sm = 0.f, dn = 0.f;
        #pragma unroll
        for (int i = 0; i < 16; ++i) { ce += red[0][i]; sm += red[1][i]; dn += red[2][i]; }
        parts[3 * (size_t)blk + 0] = ce;
        parts[3 * (size_t)blk + 1] = sm;
        parts[3 * (size_t)blk + 2] = dn;
    }
}

// Deterministic fixed-order final reduction over per-block partials.
__global__ __launch_bounds__(256)
void ce_finalize(const float* __restrict__ parts, float* __restrict__ out, int nblk)
{
    __shared__ float sce[256], ssm[256], sdn[256];
    const int t = threadIdx.x;
    float ce = 0.f, sm = 0.f, dn = 0.f;
    for (int p = t; p < nblk; p += 256) {
        ce += parts[3 * (size_t)p + 0];
        sm += parts[3 * (size_t)p + 1];
        dn += parts[3 * (size_t)p + 2];
    }
    sce[t] = ce; ssm[t] = sm; sdn[t] = dn;
    __syncthreads();
    for (int s = 128; s > 0; s >>= 1) {
        if (t < s) { sce[t] += sce[t + s]; ssm[t] += ssm[t + s]; sdn[t] += sdn[t + s]; }
        __syncthreads();
    }
    if (t == 0) {
        out[0] = sce[0] / sdn[0];   // ce_loss
        out[1] = ssm[0] / sdn[0];   // sim_loss
    }
}

extern "C" void kernel_launch(void* const* d_in, const int* in_sizes, int n_in,
                              void* d_out, int out_size, void* d_ws, size_t ws_size,
                              hipStream_t stream) {
    const float*         S   = (const float*)d_in[0];
    const float*         C   = (const float*)d_in[1];
    const unsigned char* PAD = (const unsigned char*)d_in[2];
    const unsigned char* VM  = (const unsigned char*)d_in[3];
    const int*           NEG = (const int*)d_in[4];
    float* out   = (float*)d_out;
    float* parts = (float*)d_ws;   // QTILES_ * 3 floats = 24 KB

    ce_sddmm_wmma<<<QTILES_, 256, 0, stream>>>(S, C, PAD, VM, NEG, parts);
    ce_finalize<<<1, 256, 0, stream>>>(parts, out, QTILES_);
}